// LlamaDecodeStaticCacheAttentionWrapper_49495203119708
// MI455X (gfx1250) — compile-verified
//
#include <hip/hip_runtime.h>
#include <hip/hip_bf16.h>
#include <math.h>

#define B_ 16
#define D_ 4096
#define NH_ 32
#define NKV_ 8
#define HD_ 128
#define NCHUNK 8
#define PSTRIDE (HD_ + 4)   // partial record: [m, l, pad, pad, ctx[128]] -> 16B aligned

typedef __attribute__((ext_vector_type(2))) float v2f;
typedef __attribute__((ext_vector_type(8))) float v8f;

// ---------------------------------------------------------------------------
// Out[16 x E] = A[16 x K] * W[E x K]^T  using V_WMMA_F32_16X16X4_F32.
// One wave owns one 16x16 output tile. K-reduction order is permuted so each
// lane feeds two WMMA ops from a single float4 load of A and of W:
//   half=0 lanes supply K = k..k+1 (op0) and k+2..k+3 (op1)
//   half=1 lanes supply K = k+4..k+5 (op0) and k+6..k+7 (op1)
// A and W use the same permutation, so the dot product is exact.
// ---------------------------------------------------------------------------
__global__ __launch_bounds__(64) void gemm16_wmma(const float* __restrict__ A,
                                                  const float* __restrict__ W,
                                                  float* __restrict__ Out,
                                                  int K, int E) {
  const int wave = threadIdx.x >> 5;
  const int lane = threadIdx.x & 31;
  const int tile = blockIdx.x * 2 + wave;
  if (tile * 16 >= E) return;
  const int m = lane & 15;
  const int half = lane >> 4;

  const float4* __restrict__ Arow = (const float4*)(A + (size_t)m * K);
  const float4* __restrict__ Wrow = (const float4*)(W + ((size_t)tile * 16 + m) * K);

  v8f acc = {};
  const int kq = K >> 2;  // float4s per row
#pragma unroll 4
  for (int k4 = 0; k4 < kq; k4 += 2) {
    float4 a = Arow[k4 + half];
    float4 w = Wrow[k4 + half];
    __builtin_prefetch(&Wrow[k4 + half + 64], 0, 1);  // global_prefetch_b8, 1KB ahead
    v2f a0 = {a.x, a.y}, a1 = {a.z, a.w};
    v2f w0 = {w.x, w.y}, w1 = {w.z, w.w};
    acc = __builtin_amdgcn_wmma_f32_16x16x4_f32(false, a0, false, w0, (short)0, acc,
                                                false, false);
    acc = __builtin_amdgcn_wmma_f32_16x16x4_f32(false, a1, false, w1, (short)0, acc,
                                                false, false);
  }
  // C/D layout: VGPR j -> row j (lanes 0-15) / row j+8 (lanes 16-31), col = lane%16
#pragma unroll
  for (int j = 0; j < 8; ++j)
    Out[(size_t)(j + 8 * half) * E + tile * 16 + m] = acc[j];
}

// ---------------------------------------------------------------------------
// RoPE applied in-place to q (B x 32 x 128) and k_new (B x 8 x 128) in workspace.
// One thread per rotation pair (j, j+64).
// ---------------------------------------------------------------------------
__global__ void rope_apply(float* __restrict__ q, float* __restrict__ k,
                           const long long* __restrict__ pos_ids) {
  int idx = blockIdx.x * blockDim.x + threadIdx.x;
  if (idx >= B_ * (NH_ + NKV_) * (HD_ / 2)) return;
  int j = idx & 63;
  int rest = idx >> 6;
  int h = rest % (NH_ + NKV_);
  int b = rest / (NH_ + NKV_);
  float pos = (float)pos_ids[b];
  // inv_freq = 10000^(-2j/128) = exp(-j * ln(10000)/64)
  float inv = __expf(-(float)j * (9.210340371976184f / 64.0f));
  float ang = pos * inv;
  float s, c;
  __sincosf(ang, &s, &c);
  float* base = (h < NH_) ? (q + ((size_t)b * NH_ + h) * HD_)
                          : (k + ((size_t)b * NKV_ + (h - NH_)) * HD_);
  float x1 = base[j], x2 = base[j + 64];
  base[j] = x1 * c - x2 * s;
  base[j + 64] = x2 * c + x1 * s;
}

__device__ __forceinline__ float wave_sum(float v) {
#pragma unroll
  for (int off = 16; off > 0; off >>= 1) v += __shfl_xor(v, off, 32);
  return v;
}

// ---------------------------------------------------------------------------
// Fused KV-cache concat + flash-decode partial attention.
// grid = (NCHUNK, NKV, B), block = 256 (8 waves). Each wave streams a slice of
// S: one b128 load per K row and V row (32 lanes x float4 = 128 floats = one
// row), stores it to the cache outputs, and runs online softmax for the 4
// grouped Q heads. Per-block LDS combine -> per-chunk partial (m, l, ctx).
// ---------------------------------------------------------------------------
__global__ __launch_bounds__(256) void attn_partial(
    const float* __restrict__ past_k, const float* __restrict__ past_v,
    const float* __restrict__ k_new, const float* __restrict__ v_new,
    const float* __restrict__ q, const float* __restrict__ amask,
    float* __restrict__ k_cache, float* __restrict__ v_cache,
    float* __restrict__ parts, int S, int PAST, int CS) {
  const int b = blockIdx.z, kvh = blockIdx.y, chunk = blockIdx.x;
  const int wave = threadIdx.x >> 5, lane = threadIdx.x & 31;
  const int sw = (CS + 7) >> 3;
  int s0 = chunk * CS + wave * sw;
  int s1 = s0 + sw;
  int cend = chunk * CS + CS;
  if (s1 > cend) s1 = cend;
  if (s1 > S) s1 = S;

  float4 qv[4];
  const float* qb = q + ((size_t)b * NH_ + (size_t)kvh * 4) * HD_ + 4 * lane;
#pragma unroll
  for (int i = 0; i < 4; ++i) qv[i] = *(const float4*)(qb + (size_t)i * HD_);

  const float4* Kp = (const float4*)(past_k + (size_t)(b * NKV_ + kvh) * PAST * HD_);
  const float4* Vp = (const float4*)(past_v + (size_t)(b * NKV_ + kvh) * PAST * HD_);
  const float4* Kn = (const float4*)(k_new + (size_t)(b * NKV_ + kvh) * HD_);
  const float4* Vn = (const float4*)(v_new + (size_t)(b * NKV_ + kvh) * HD_);
  float4* Ko = (float4*)(k_cache + (size_t)(b * NKV_ + kvh) * S * HD_);
  float4* Vo = (float4*)(v_cache + (size_t)(b * NKV_ + kvh) * S * HD_);
  const float* mrow = amask + (size_t)b * S;

  float mi[4] = {-INFINITY, -INFINITY, -INFINITY, -INFINITY};
  float li[4] = {0.f, 0.f, 0.f, 0.f};
  float4 acc[4];
#pragma unroll
  for (int i = 0; i < 4; ++i) acc[i] = make_float4(0.f, 0.f, 0.f, 0.f);

  for (int s = s0; s < s1; ++s) {
    float4 kk, vv;
    if (s < PAST) {
      kk = Kp[(size_t)s * 32 + lane];
      vv = Vp[(size_t)s * 32 + lane];
      if (s + 16 < PAST) {
        __builtin_prefetch(&Kp[(size_t)(s + 16) * 32 + lane], 0, 1);
        __builtin_prefetch(&Vp[(size_t)(s + 16) * 32 + lane], 0, 1);
      }
    } else {  // the freshly projected token (s == PAST, T == 1)
      kk = Kn[lane];
      vv = Vn[lane];
    }
    Ko[(size_t)s * 32 + lane] = kk;  // cache concat output, written exactly once
    Vo[(size_t)s * 32 + lane] = vv;
    float msk = mrow[s];
#pragma unroll
    for (int qh = 0; qh < 4; ++qh) {
      float p = kk.x * qv[qh].x + kk.y * qv[qh].y + kk.z * qv[qh].z + kk.w * qv[qh].w;
      p = wave_sum(p);
      float sc = p * 0.08838834764831845f + msk;  // 1/sqrt(128)
      float nm = fmaxf(mi[qh], sc);
      float corr = __expf(mi[qh] - nm);  // exp(-inf)=0 on first step
      float pe = __expf(sc - nm);
      li[qh] = li[qh] * corr + pe;
      acc[qh].x = acc[qh].x * corr + pe * vv.x;
      acc[qh].y = acc[qh].y * corr + pe * vv.y;
      acc[qh].z = acc[qh].z * corr + pe * vv.z;
      acc[qh].w = acc[qh].w * corr + pe * vv.w;
      mi[qh] = nm;
    }
  }

  __shared__ float sm[8][4];
  __shared__ float sl[8][4];
  __shared__ float sctx[8][4][HD_];
#pragma unroll
  for (int qh = 0; qh < 4; ++qh) {
    if (lane == 0) { sm[wave][qh] = mi[qh]; sl[wave][qh] = li[qh]; }
    ((float4*)sctx[wave][qh])[lane] = acc[qh];
  }
  __syncthreads();
  if (wave < 4) {
    const int qh = wave;
    float M = -INFINITY;
#pragma unroll
    for (int w = 0; w < 8; ++w) M = fmaxf(M, sm[w][qh]);
    float L = 0.f;
    float4 C = make_float4(0.f, 0.f, 0.f, 0.f);
#pragma unroll
    for (int w = 0; w < 8; ++w) {
      float mw = sm[w][qh];
      float cw = (mw == -INFINITY) ? 0.f : __expf(mw - M);
      L += cw * sl[w][qh];
      float4 t = ((const float4*)sctx[w][qh])[lane];
      C.x += cw * t.x; C.y += cw * t.y; C.z += cw * t.z; C.w += cw * t.w;
    }
    float* P = parts + (((size_t)chunk * B_ + b) * NH_ + kvh * 4 + qh) * PSTRIDE;
    if (lane == 0) { P[0] = M; P[1] = L; }
    ((float4*)(P + 4))[lane] = C;
  }
}

// ---------------------------------------------------------------------------
// Combine the NCHUNK partials per (b, head) -> ctx[b][h*128 + d].
// One wave per (b, head).
// ---------------------------------------------------------------------------
__global__ __launch_bounds__(256) void attn_reduce(const float* __restrict__ parts,
                                                   float* __restrict__ ctx) {
  const int wave = threadIdx.x >> 5, lane = threadIdx.x & 31;
  const int wid = blockIdx.x * 8 + wave;  // b*NH_ + h
  if (wid >= B_ * NH_) return;
  float M = -INFINITY;
#pragma unroll
  for (int c = 0; c < NCHUNK; ++c)
    M = fmaxf(M, parts[((size_t)c * B_ * NH_ + wid) * PSTRIDE]);
  float L = 0.f;
  float4 C = make_float4(0.f, 0.f, 0.f, 0.f);
#pragma unroll
  for (int c = 0; c < NCHUNK; ++c) {
    const float* P = parts + ((size_t)c * B_ * NH_ + wid) * PSTRIDE;
    float mw = P[0];
    float cw = (mw == -INFINITY) ? 0.f : __expf(mw - M);
    L += cw * P[1];
    float4 t = ((const float4*)(P + 4))[lane];
    C.x += cw * t.x; C.y += cw * t.y; C.z += cw * t.z; C.w += cw * t.w;
  }
  float inv = 1.f / L;
  float4 o = make_float4(C.x * inv, C.y * inv, C.z * inv, C.w * inv);
  *(float4*)(ctx + (size_t)wid * HD_ + 4 * lane) = o;
}

// ---------------------------------------------------------------------------
extern "C" void kernel_launch(void* const* d_in, const int* in_sizes, int n_in,
                              void* d_out, int out_size, void* d_ws, size_t ws_size,
                              hipStream_t stream) {
  const float* hs       = (const float*)d_in[0];
  const float* amask    = (const float*)d_in[1];
  const long long* pos  = (const long long*)d_in[2];
  const float* past_k   = (const float*)d_in[3];
  const float* past_v   = (const float*)d_in[4];
  const float* Wq       = (const float*)d_in[5];
  const float* Wk       = (const float*)d_in[6];
  const float* Wv       = (const float*)d_in[7];
  const float* Wo       = (const float*)d_in[8];
  const int PAST = in_sizes[3] / (B_ * NKV_ * HD_);
  const int S = PAST + 1;

  // d_out = [attn_output (B*4096) | k_cache (B*8*S*128) | v_cache (B*8*S*128)]
  float* attn_out = (float*)d_out;
  float* k_cache = attn_out + (size_t)B_ * NH_ * HD_;
  float* v_cache = k_cache + (size_t)B_ * NKV_ * S * HD_;

  // workspace: q | k_new | v_new | ctx | partials
  float* q     = (float*)d_ws;
  float* k_new = q + (size_t)B_ * NH_ * HD_;
  float* v_new = k_new + (size_t)B_ * NKV_ * HD_;
  float* ctx   = v_new + (size_t)B_ * NKV_ * HD_;
  float* parts = ctx + (size_t)B_ * NH_ * HD_;

  // QKV projections: M=16 WMMA GEMMs (2 waves / 64-thread block, 1 tile each)
  gemm16_wmma<<<dim3((NH_ * HD_ / 16) / 2), 64, 0, stream>>>(hs, Wq, q, D_, NH_ * HD_);
  gemm16_wmma<<<dim3((NKV_ * HD_ / 16) / 2), 64, 0, stream>>>(hs, Wk, k_new, D_, NKV_ * HD_);
  gemm16_wmma<<<dim3((NKV_ * HD_ / 16) / 2), 64, 0, stream>>>(hs, Wv, v_new, D_, NKV_ * HD_);

  // RoPE on q and k_new (workspace, inputs untouched)
  {
    int tot = B_ * (NH_ + NKV_) * (HD_ / 2);
    rope_apply<<<(tot + 255) / 256, 256, 0, stream>>>(q, k_new, pos);
  }

  // Fused cache-concat + flash-decode partials
  int CS = (S + NCHUNK - 1) / NCHUNK;
  attn_partial<<<dim3(NCHUNK, NKV_, B_), 256, 0, stream>>>(
      past_k, past_v, k_new, v_new, q, amask, k_cache, v_cache, parts, S, PAST, CS);
  attn_reduce<<<(B_ * NH_ + 7) / 8, 256, 0, stream>>>(parts, ctx);

  // Output projection: attn_out[b][d] = sum_e ctx[b][e] * Wo[d][e]
  gemm16_wmma<<<dim3((D_ / 16) / 2), 64, 0, stream>>>(ctx, Wo, attn_out, NH_ * HD_, D_);
}